// TopK_59854664237629
// MI455X (gfx1250) — compile-verified
//
#include <hip/hip_runtime.h>
#include <stdint.h>

typedef __attribute__((ext_vector_type(2))) float v2f;
typedef __attribute__((ext_vector_type(8))) float v8f;

#define SORT_N 4096

// Monotone map: float -> uint32 preserving order (larger float => larger key)
static __device__ __forceinline__ uint32_t mono_key(float f) {
  uint32_t u = __float_as_uint(f);
  return (u & 0x80000000u) ? ~u : (u | 0x80000000u);
}

// ---------------- init: zero histograms + ctrl ----------------
__global__ __launch_bounds__(256)
void init_kernel(uint32_t* __restrict__ hist, uint32_t* __restrict__ ctrl, int k) {
  int i = blockIdx.x * blockDim.x + threadIdx.x;
  if (i < 3 * 2048) hist[i] = 0u;
  if (i == 0) { ctrl[0] = (uint32_t)k; ctrl[1] = 0u; ctrl[2] = 0u; }
}

// ---------------- scores via V_WMMA_F32_16X16X4_F32 ----------------
// One wave handles a 16-row tile. A(16x4 f32): lane m (0-15) holds row m,
// VGPR j / lane-half h -> K = j + 2h (ISA 7.12.2, 32-bit A 16x4 table).
// B(4x16): scorer chunk replicated across all 16 columns with the same
// K = j + 2h mapping -> every column of D = scores for the 16 rows.
__global__ __launch_bounds__(256)
void scores_wmma_kernel(const float* __restrict__ emb,
                        const float* __restrict__ maskv,
                        const float* __restrict__ scorer,
                        uint32_t* __restrict__ mkey,
                        int F, int tiles) {
  const int lane  = threadIdx.x & 31;
  const int half  = lane >> 4;     // 0 or 1
  const int mrow  = lane & 15;     // row within tile for A operand
  const int gwave = blockIdx.x * (blockDim.x >> 5) + (threadIdx.x >> 5);
  const int nwav  = gridDim.x * (blockDim.x >> 5);

  // sum(scorer), wave-redundant
  float s = 0.f;
  for (int i = lane; i < F; i += 32) s += scorer[i];
#pragma unroll
  for (int off = 16; off > 0; off >>= 1) s += __shfl_xor(s, off, 32);

  for (int tile = gwave; tile < tiles; tile += nwav) {
    const int rowBase = tile * 16;
    const float* arow = emb + (size_t)(rowBase + mrow) * F + half * 2;
    const float* brow = scorer + half * 2;
    v8f acc = {0.f, 0.f, 0.f, 0.f, 0.f, 0.f, 0.f, 0.f};
    for (int k0 = 0; k0 < F; k0 += 4) {
      float2 av = *(const float2*)(arow + k0);
      float2 bv = *(const float2*)(brow + k0);
      v2f a; a.x = av.x; a.y = av.y;
      v2f b; b.x = bv.x; b.y = bv.y;
      acc = __builtin_amdgcn_wmma_f32_16x16x4_f32(false, a, false, b,
                                                  (short)0, acc, false, false);
    }
    // D layout: lane n<16: VGPR r -> (M=r, N=n); lane n>=16: (M=8+r, N=n-16).
    float vals[8];
#pragma unroll
    for (int i = 0; i < 8; ++i) vals[i] = acc[i];
    if (mrow < 8) {
      int r   = (half << 3) + mrow;      // 0..15
      int row = rowBase + r;
      float sc = vals[mrow] / s + maskv[row];
      mkey[row] = mono_key(sc);
    }
  }
}

// Tail rows (N % 16), plain VALU dot (not launched when N % 16 == 0)
__global__ __launch_bounds__(64)
void scores_tail_kernel(const float* __restrict__ emb,
                        const float* __restrict__ maskv,
                        const float* __restrict__ scorer,
                        uint32_t* __restrict__ mkey,
                        int N, int F, int startRow) {
  int row = startRow + blockIdx.x * blockDim.x + threadIdx.x;
  if (row >= N) return;
  float s = 0.f, d = 0.f;
  for (int i = 0; i < F; ++i) { s += scorer[i]; d += emb[(size_t)row * F + i] * scorer[i]; }
  mkey[row] = mono_key(d / s + maskv[row]);
}

// ---------------- radix-select level: histogram ----------------
__global__ __launch_bounds__(256)
void hist_kernel(const uint32_t* __restrict__ mkey, int N,
                 uint32_t* __restrict__ hist, const uint32_t* __restrict__ ctrl,
                 int prefixShift, int binShift, uint32_t binMask) {
  __shared__ uint32_t lh[2048];
  for (int i = threadIdx.x; i < 2048; i += blockDim.x) lh[i] = 0u;
  __syncthreads();
  uint32_t pref = (prefixShift < 32) ? (ctrl[1] >> prefixShift) : 0u;
  for (int i = blockIdx.x * blockDim.x + threadIdx.x; i < N;
       i += gridDim.x * blockDim.x) {
    uint32_t v = mkey[i];
    if (prefixShift >= 32 || (v >> prefixShift) == pref)
      atomicAdd(&lh[(v >> binShift) & binMask], 1u);
  }
  __syncthreads();
  for (int i = threadIdx.x; i < 2048; i += blockDim.x) {
    uint32_t c = lh[i];
    if (c) atomicAdd(&hist[i], c);
  }
}

// ---------------- radix-select level: pick bin (1 block, 256 thr) ----------------
__global__ __launch_bounds__(256)
void pick_kernel(const uint32_t* __restrict__ hist, uint32_t* __restrict__ ctrl,
                 int nbins, int binShift) {
  __shared__ uint32_t chunkSum[256];
  __shared__ uint32_t chunkAbove[256];
  const int t = threadIdx.x;
  const int chunk = nbins >> 8;       // 8 (2048 bins) or 4 (1024 bins)
  uint32_t h[8];
  uint32_t sum = 0;
  for (int j = 0; j < chunk; ++j) { h[j] = hist[t * chunk + j]; sum += h[j]; }
  chunkSum[t] = sum;
  uint32_t kRem = ctrl[0];            // read before any write below
  __syncthreads();
  if (t == 0) {
    uint32_t run = 0;
    for (int c = 255; c >= 0; --c) { chunkAbove[c] = run; run += chunkSum[c]; }
  }
  __syncthreads();
  uint32_t cum = chunkAbove[t];
  for (int j = chunk - 1; j >= 0; --j) {
    uint32_t above = cum;
    if (kRem > above && kRem <= above + h[j]) {   // exactly one winner
      ctrl[0] = kRem - above;
      ctrl[1] = ctrl[1] | ((uint32_t)(t * chunk + j) << binShift);
    }
    cum += h[j];
  }
}

// ---------------- compact all keys >= threshold ----------------
__global__ __launch_bounds__(256)
void compact_kernel(const uint32_t* __restrict__ mkey, int N,
                    uint32_t* __restrict__ ctrl,
                    unsigned long long* __restrict__ cand) {
  uint32_t thr = ctrl[1];
  for (int i = blockIdx.x * blockDim.x + threadIdx.x; i < N;
       i += gridDim.x * blockDim.x) {
    uint32_t v = mkey[i];
    if (v >= thr) {
      uint32_t pos = atomicAdd(&ctrl[2], 1u);
      if (pos < SORT_N)
        cand[pos] = ((unsigned long long)v << 32) | (uint32_t)(~(uint32_t)i);
    }
  }
}

// ---------------- bitonic sort (desc) of <=4096 keys, take top-k indices ----------------
__global__ __launch_bounds__(1024)
void sort_kernel(const unsigned long long* __restrict__ cand,
                 const uint32_t* __restrict__ ctrl,
                 uint32_t* __restrict__ topIdx, int k) {
  __shared__ unsigned long long key[SORT_N];
  const int t = threadIdx.x;
  uint32_t cnt = ctrl[2]; if (cnt > SORT_N) cnt = SORT_N;
  for (int i = t; i < SORT_N; i += 1024)
    key[i] = (i < (int)cnt) ? cand[i] : 0ull;   // real keys are > 0
  __syncthreads();
  for (int kk = 2; kk <= SORT_N; kk <<= 1) {
    for (int j = kk >> 1; j > 0; j >>= 1) {
      for (int i = t; i < SORT_N; i += 1024) {
        int ixj = i ^ j;
        if (ixj > i) {
          unsigned long long a = key[i], b = key[ixj];
          bool up = ((i & kk) == 0);
          if (up ? (a < b) : (a > b)) { key[i] = b; key[ixj] = a; }
        }
      }
      __syncthreads();
    }
  }
  for (int i = t; i < k && i < SORT_N; i += 1024)
    topIdx[i] = ~((uint32_t)(key[i] & 0xFFFFFFFFull));
}

// ---------------- gather + transpose: out[f*k + j] = emb[topIdx[j]][f] ----------------
__global__ __launch_bounds__(256)
void gather_kernel(const float* __restrict__ emb,
                   const uint32_t* __restrict__ topIdx,
                   float* __restrict__ out, int N, int F, int k) {
  __shared__ float tile[32][257];
  const int j0 = blockIdx.x * 32;
  const int t  = threadIdx.x;
  const int FT = (F < 256) ? F : 256;
  for (int e = t; e < 32 * FT; e += 256) {
    int r = e / FT, f = e - r * FT;
    int j = j0 + r;
    if (j < k) {
      uint32_t idx = topIdx[j];
      if (idx >= (uint32_t)N) idx = 0u;
      tile[r][f] = emb[(size_t)idx * F + f];
    }
  }
  __syncthreads();
  for (int e = t; e < FT * 32; e += 256) {
    int f = e >> 5, jj = e & 31;
    if (j0 + jj < k) out[(size_t)f * k + (j0 + jj)] = tile[jj][f];
  }
}

extern "C" void kernel_launch(void* const* d_in, const int* in_sizes, int n_in,
                              void* d_out, int out_size, void* d_ws, size_t ws_size,
                              hipStream_t stream) {
  const float* emb    = (const float*)d_in[0];
  const float* maskv  = (const float*)d_in[1];
  const float* scorer = (const float*)d_in[2];
  const int F = in_sizes[2];          // 256
  const int N = in_sizes[0] / F;      // 100000
  const int k = out_size / F;         // 2000
  float* out = (float*)d_out;

  // workspace layout
  uint8_t* w = (uint8_t*)d_ws;
  uint32_t* mkey = (uint32_t*)w;
  size_t off = (((size_t)N * 4) + 255) & ~(size_t)255;
  uint32_t* hist = (uint32_t*)(w + off); off += 3 * 2048 * 4;   // hist0|hist1|hist2
  uint32_t* ctrl = (uint32_t*)(w + off); off += 64;             // kRem, prefix, candCnt
  uint32_t* topIdx = (uint32_t*)(w + off); off += SORT_N * 4;
  off = (off + 7) & ~(size_t)7;
  unsigned long long* cand = (unsigned long long*)(w + off);

  init_kernel<<<24, 256, 0, stream>>>(hist, ctrl, k);

  const int tiles = N / 16;
  if (tiles > 0) {
    int blocks = (tiles + 7) / 8;     // 8 waves per 256-thread block
    scores_wmma_kernel<<<blocks, 256, 0, stream>>>(emb, maskv, scorer, mkey, F, tiles);
  }
  const int rem = N % 16;
  if (rem > 0)
    scores_tail_kernel<<<1, 64, 0, stream>>>(emb, maskv, scorer, mkey, N, F, tiles * 16);

  // 3-level radix select on 32-bit monotone keys: bits [31:21],[20:10],[9:0]
  hist_kernel<<<160, 256, 0, stream>>>(mkey, N, hist,        ctrl, 32, 21, 2047u);
  pick_kernel<<<1, 256, 0, stream>>>(hist,        ctrl, 2048, 21);
  hist_kernel<<<160, 256, 0, stream>>>(mkey, N, hist + 2048, ctrl, 21, 10, 2047u);
  pick_kernel<<<1, 256, 0, stream>>>(hist + 2048, ctrl, 2048, 10);
  hist_kernel<<<160, 256, 0, stream>>>(mkey, N, hist + 4096, ctrl, 10,  0, 1023u);
  pick_kernel<<<1, 256, 0, stream>>>(hist + 4096, ctrl, 1024,  0);

  compact_kernel<<<160, 256, 0, stream>>>(mkey, N, ctrl, cand);
  sort_kernel<<<1, 1024, 0, stream>>>(cand, ctrl, topIdx, k);
  gather_kernel<<<(k + 31) / 32, 256, 0, stream>>>(emb, topIdx, out, N, F, k);
}